// PPS_loss_86191403696339
// MI455X (gfx1250) — compile-verified
//
#include <hip/hip_runtime.h>
#include <hip/hip_bf16.h>
#include <math.h>

#define EPSF 0.001f
#define MUF  1.2f

// Tile: 32 x 16 output pixels per block, 256 threads (8 wave32), 2 px/thread.
#define TLX 32
#define TLY 16
#define PITCH (TLX + 2)   // 34
#define ROWS  (TLY + 2)   // 18
#define NELEM (PITCH * ROWS) // 612

// ---------------------------------------------------------------------------
// small float3 helpers
// ---------------------------------------------------------------------------
struct F3 { float x, y, z; };

__device__ __forceinline__ F3 crossf(F3 u, F3 v) {
    return { u.y * v.z - u.z * v.y,
             u.z * v.x - u.x * v.z,
             u.x * v.y - u.y * v.x };
}
__device__ __forceinline__ F3 addf(F3 u, F3 v) {
    return { u.x + v.x, u.y + v.y, u.z + v.z };
}
__device__ __forceinline__ F3 normeps(F3 v) {
    float n   = sqrtf(v.x * v.x + v.y * v.y + v.z * v.z);
    float inv = 1.0f / (n + EPSF);
    return { v.x * inv, v.y * inv, v.z * inv };
}

// ---------------------------------------------------------------------------
// Per-pixel PPS term.  t = LDS depth tile (with 1-halo), (ly,lx) = center.
// a* = (x-cx)/fx for x-1,x,x+1 ; b* = (y-cy)/fy for y-1,y,y+1.
// interior==false -> normal = (0,0,1) per reference init.
// ---------------------------------------------------------------------------
__device__ __forceinline__ float compute_pps(const float (*t)[PITCH], int ly, int lx,
                                             float am, float a, float ap,
                                             float bm, float bv, float bp,
                                             bool interior)
{
    float dC = t[ly][lx];
    // point p = (a*d, b*d, d); points_p = (a, b, 1); dvec = p - points_p
    float px = a * dC, py = bv * dC, pz = dC;
    float dx = px - a, dy = py - bv, dz = pz - 1.0f;
    float r2   = dx * dx + dy * dy + dz * dz;
    float r    = sqrtf(r2);
    float invr = 1.0f / r;
    float Lx = dx * invr, Ly = dy * invr, Lz = dz * invr;
    // A = Lz^MU / r^2  (Lz > 0 because depth > 1)
    float A = exp2f(MUF * log2f(Lz)) * (1.0f / r2);

    float Nx = 0.0f, Ny = 0.0f, Nz = 1.0f;
    if (interior) {
        float dN  = t[ly - 1][lx    ];
        float dNE = t[ly - 1][lx + 1];
        float dE  = t[ly    ][lx + 1];
        float dS  = t[ly + 1][lx    ];
        float dSW = t[ly + 1][lx - 1];
        float dW  = t[ly    ][lx - 1];

        F3 vN  = { a  * dN  - px, bm * dN  - py, dN  - pz };
        F3 vNE = { ap * dNE - px, bm * dNE - py, dNE - pz };
        F3 vE  = { ap * dE  - px, bv * dE  - py, dE  - pz };
        F3 vS  = { a  * dS  - px, bp * dS  - py, dS  - pz };
        F3 vSW = { am * dSW - px, bp * dSW - py, dSW - pz };
        F3 vW  = { am * dW  - px, bv * dW  - py, dW  - pz };

        F3 s = normeps(crossf(vE,  vNE));
        s = addf(s, normeps(crossf(vNE, vN)));
        s = addf(s, normeps(crossf(vN,  vW)));
        s = addf(s, normeps(crossf(vW,  vSW)));
        s = addf(s, normeps(crossf(vSW, vS)));
        s = addf(s, normeps(crossf(vS,  vE)));

        float mx = s.x * (1.0f / 6.0f);
        float my = s.y * (1.0f / 6.0f);
        float mz = s.z * (1.0f / 6.0f);
        float mn  = sqrtf(mx * mx + my * my + mz * mz);
        float inv = 1.0f / (mn + EPSF);
        Nx = mx * inv; Ny = my * inv; Nz = mz * inv;
    }
    return A * (Lx * Nx + Ly * Ny + Lz * Nz);
}

// ---------------------------------------------------------------------------
// async global -> LDS staging (CDNA5 GLOBAL_LOAD_ASYNC_TO_LDS_B32, ASYNCcnt)
// lds_off: 32-bit LDS byte offset (low 32 bits of the flat shared pointer)
// ---------------------------------------------------------------------------
__device__ __forceinline__ void async_load_f32(unsigned lds_off, const float* gptr)
{
    asm volatile("global_load_async_to_lds_b32 %0, %1, off"
                 :: "v"(lds_off), "v"(gptr) : "memory");
}
__device__ __forceinline__ void wait_async_all()
{
    asm volatile("s_wait_asynccnt 0" ::: "memory");
}

// ---------------------------------------------------------------------------
// Main kernel: one block = 32x16 output pixels of one image (2 px / thread).
// ---------------------------------------------------------------------------
__global__ __launch_bounds__(256)
void pps_loss_kernel(const float* __restrict__ depth,
                     const float* __restrict__ depth_gt,
                     const float* __restrict__ mask,
                     const float* __restrict__ K,
                     float* __restrict__ accum,
                     int H, int W)
{
    __shared__ float tP[ROWS][PITCH];   // pred depth tile + halo
    __shared__ float tG[ROWS][PITCH];   // gt depth tile + halo
    __shared__ float red[8];            // per-wave partials

    const int tx  = threadIdx.x;            // 0..31
    const int ty  = threadIdx.y;            // 0..7
    const int tid = ty * 32 + tx;           // 0..255
    const int bx0 = blockIdx.x * TLX;
    const int by0 = blockIdx.y * TLY;
    const int b   = blockIdx.z;

    const float fx = K[0], cx = K[2], fy = K[4], cy = K[5];

    const size_t imgOff = (size_t)b * H * W;
    const float* dP = depth    + imgOff;
    const float* dG = depth_gt + imgOff;

    // ---- stage tiles into LDS with async copies (clamped halo) ----
    for (int i = tid; i < NELEM; i += 256) {
        int ly = i / PITCH, lx = i - ly * PITCH;
        int gy = by0 - 1 + ly; gy = gy < 0 ? 0 : (gy > H - 1 ? H - 1 : gy);
        int gx = bx0 - 1 + lx; gx = gx < 0 ? 0 : (gx > W - 1 ? W - 1 : gx);
        const int idx = gy * W + gx;
        async_load_f32((unsigned)(size_t)&tP[ly][lx], dP + idx);
        async_load_f32((unsigned)(size_t)&tG[ly][lx], dG + idx);
    }
    wait_async_all();
    __syncthreads();

    // ---- per-pixel PPS for pred and gt (2 rows per thread) ----
    const int   gx = bx0 + tx;
    const int   lx = tx + 1;
    const float a  = ((float)gx       - cx) / fx;
    const float am = ((float)(gx - 1) - cx) / fx;
    const float ap = ((float)(gx + 1) - cx) / fx;
    const bool  xin = (gx > 0) && (gx < W - 1);

    float val = 0.0f;
    #pragma unroll
    for (int sub = 0; sub < 2; ++sub) {
        const int gy = by0 + ty + sub * 8;
        const int ly = ty + 1 + sub * 8;
        const bool interior = xin && (gy > 0) && (gy < H - 1);

        const float bv = ((float)gy       - cy) / fy;
        const float bm = ((float)(gy - 1) - cy) / fy;
        const float bp = ((float)(gy + 1) - cy) / fy;

        float pps_p = compute_pps(tP, ly, lx, am, a, ap, bm, bv, bp, interior);
        float pps_g = compute_pps(tG, ly, lx, am, a, ap, bm, bv, bp, interior);

        // mask is streamed exactly once -> non-temporal, don't pollute L2
        float m = __builtin_nontemporal_load(mask + imgOff + (size_t)gy * W + gx);
        val += fabsf(m * (pps_p - pps_g));
    }

    // ---- reduction: wave32 shuffle, then 8 partials via LDS, one atomic ----
    #pragma unroll
    for (int off = 16; off > 0; off >>= 1)
        val += __shfl_down(val, off, 32);
    if ((tid & 31) == 0) red[tid >> 5] = val;
    __syncthreads();
    if (tid == 0) {
        float s = 0.0f;
        #pragma unroll
        for (int i = 0; i < 8; ++i) s += red[i];
        atomicAdd(accum, s);
    }
}

__global__ void pps_zero_kernel(float* acc) { acc[0] = 0.0f; }

__global__ void pps_finalize_kernel(const float* __restrict__ acc,
                                    float* __restrict__ out, float inv_n)
{
    out[0] = acc[0] * inv_n;
}

// ---------------------------------------------------------------------------
// launch
// ---------------------------------------------------------------------------
extern "C" void kernel_launch(void* const* d_in, const int* in_sizes, int n_in,
                              void* d_out, int out_size, void* d_ws, size_t ws_size,
                              hipStream_t stream)
{
    const float* depth    = (const float*)d_in[0];
    const float* depth_gt = (const float*)d_in[1];
    const float* mask     = (const float*)d_in[2];
    const float* K        = (const float*)d_in[3];
    float* out = (float*)d_out;
    float* acc = (float*)d_ws;

    const int H = 480, W = 640;
    const int B = in_sizes[0] / (H * W);   // 8 for the reference setup

    pps_zero_kernel<<<1, 1, 0, stream>>>(acc);

    dim3 blk(32, 8, 1);
    dim3 grd(W / TLX, H / TLY, B);         // 20 x 30 x 8 = 4800 blocks
    pps_loss_kernel<<<grd, blk, 0, stream>>>(depth, depth_gt, mask, K, acc, H, W);

    const float inv_n = 1.0f / (float)((size_t)B * H * W);
    pps_finalize_kernel<<<1, 1, 0, stream>>>(acc, out, inv_n);
}